// DroneGNN_31387620999361
// MI455X (gfx1250) — compile-verified
//
#include <hip/hip_runtime.h>
#include <hip/hip_bf16.h>

typedef __attribute__((ext_vector_type(2))) float v2f;
typedef __attribute__((ext_vector_type(8))) float v8f;

#define NN 50000  // == 3125 * 16, so M-tiling needs no bounds checks

// ---------------- normalization kernels ----------------

__global__ __launch_bounds__(256) void k_deg_init(float* __restrict__ deg, int n) {
  int i = blockIdx.x * blockDim.x + threadIdx.x;
  if (i < n) deg[i] = 1.0f;  // self-loop contributes 1 to every node's degree
}

__global__ __launch_bounds__(256) void k_deg_count(const int* __restrict__ dst,
                                                   float* __restrict__ deg, int e) {
  int i = blockIdx.x * blockDim.x + threadIdx.x;
  if (i < e) atomicAdd(&deg[dst[i]], 1.0f);
}

__global__ __launch_bounds__(256) void k_dinv(float* __restrict__ deg, int n) {
  int i = blockIdx.x * blockDim.x + threadIdx.x;
  if (i < n) {
    float d = deg[i];
    deg[i] = (d > 0.0f) ? rsqrtf(d) : 0.0f;  // in-place deg -> d^{-1/2}
  }
}

__global__ __launch_bounds__(256) void k_norm(const int* __restrict__ src,
                                              const int* __restrict__ dst,
                                              const float* __restrict__ dinv,
                                              float* __restrict__ norm, int e) {
  int i = blockIdx.x * blockDim.x + threadIdx.x;
  if (i < e) norm[i] = dinv[src[i]] * dinv[dst[i]];
}

// Wt[n, k] = W[k, n]  (makes B-fragment loads contiguous float2 per lane)
template <int K, int OUTD>
__global__ __launch_bounds__(256) void k_transpose(const float* __restrict__ W,
                                                   float* __restrict__ Wt) {
  int i = blockIdx.x * blockDim.x + threadIdx.x;
  if (i < K * OUTD) {
    int k = i / OUTD;       // constant OUTD -> shift
    int n = i - k * OUTD;
    Wt[n * K + k] = W[i];
  }
}

// ---------------- dense GEMM via V_WMMA_F32_16X16X4_F32 ----------------
// H = X @ W, with W pre-transposed (Wt[n][k]).  One wave per 16x16 output
// tile; OUTD/16 waves per block cover a 16 x OUTD stripe; grid.x = NN/16.
// X tile (16 x K) is staged once per block in LDS (row stride 132 floats:
// 132 mod 64 == 4, so each wave's 64 touched dwords hit 64 distinct banks).
// Fragment layouts (ISA 7.12.2, f32):
//   A 16x4:  lane = m + 16*(k>=2), vgpr = k&1  -> contiguous float2 per lane
//   B 4x16:  lane = n + 16*(k>=2), vgpr = k&1  -> contiguous float2 from Wt
//   C/D:     vgpr r -> M = r + 8*(lane>=16), N = lane&15
template <int K, int OUTD>
__global__ __launch_bounds__(256) void k_gemm_wmma(const float* __restrict__ X,
                                                   const float* __restrict__ Wt,
                                                   float* __restrict__ H) {
  static_assert(K == 128, "K fixed at 128");
  constexpr int PAD = K + 4;  // 132
  __shared__ float xs[16 * PAD];

  const int tid   = threadIdx.x;
  const int lane  = tid & 31;
  const int wave  = tid >> 5;  // n-tile index
  const int mtile = blockIdx.x;

  // Cooperative stage of the 16 x K X tile (float4 slots; K/4 = 32 per row).
  constexpr int NTHREADS = (OUTD / 16) * 32;
  constexpr int NSLOTS   = 16 * (K / 4);  // 512
#pragma unroll
  for (int s = tid; s < NSLOTS; s += NTHREADS) {
    int row = s >> 5;
    int c4  = s & 31;
    float4 v = *(const float4*)(X + (size_t)(mtile * 16 + row) * K + c4 * 4);
    *(float4*)(&xs[row * PAD + c4 * 4]) = v;
  }
  __syncthreads();

  const int idx16 = lane & 15;
  const int half  = lane >> 4;  // 0 -> K{0,1}, 1 -> K{2,3}
  const float* xsrow = &xs[idx16 * PAD + 2 * half];
  const float* wrow  = Wt + (size_t)(wave * 16 + idx16) * K + 2 * half;

  v8f acc = {};
#pragma unroll
  for (int k0 = 0; k0 < K; k0 += 4) {
    v2f a = *(const v2f*)(xsrow + k0);  // ds_load_b64, bank-conflict-free
    v2f b = *(const v2f*)(wrow + k0);   // global_load_b64, contiguous
    acc = __builtin_amdgcn_wmma_f32_16x16x4_f32(
        /*neg_a=*/false, a, /*neg_b=*/false, b,
        /*c_mod=*/(short)0, acc, /*reuse_a=*/false, /*reuse_b=*/false);
  }

  float* hp = H + (size_t)(mtile * 16 + half * 8) * OUTD + wave * 16 + idx16;
#pragma unroll
  for (int r = 0; r < 8; ++r) hp[r * OUTD] = acc[r];
}

// ---------------- aggregation kernels ----------------

// agg[i, f] = h[i, f] * dinv[i]^2   (self-loop message; also initializes agg)
template <int OUTD>
__global__ __launch_bounds__(256) void k_self_init(const float* __restrict__ H,
                                                   const float* __restrict__ dinv,
                                                   float* __restrict__ agg,
                                                   long long total) {
  constexpr int SHIFT = (OUTD == 128) ? 7 : 6;
  long long idx = (long long)blockIdx.x * blockDim.x + threadIdx.x;
  if (idx >= total) return;
  float di = dinv[(int)(idx >> SHIFT)];
  agg[idx] = H[idx] * di * di;
}

// One wave per edge; each lane scatters OUTD/32 consecutive features.
template <int OUTD>
__global__ __launch_bounds__(256) void k_edge_scatter(const int* __restrict__ src,
                                                      const int* __restrict__ dst,
                                                      const float* __restrict__ norm,
                                                      const float* __restrict__ H,
                                                      float* __restrict__ agg, int E) {
  constexpr int VEC = OUTD / 32;  // 4 (outd=128) or 2 (outd=64)
  int e = blockIdx.x * (blockDim.x >> 5) + (threadIdx.x >> 5);
  if (e >= E) return;
  int lane = threadIdx.x & 31;
  int s = src[e];
  int d = dst[e];
  float w = norm[e];
  const float* hp = H + (size_t)s * OUTD + lane * VEC;
  float* ap = agg + (size_t)d * OUTD + lane * VEC;
  if constexpr (VEC == 4) {
    float4 v = *(const float4*)hp;
    atomicAdd(ap + 0, v.x * w);
    atomicAdd(ap + 1, v.y * w);
    atomicAdd(ap + 2, v.z * w);
    atomicAdd(ap + 3, v.w * w);
  } else {
    float2 v = *(const float2*)hp;
    atomicAdd(ap + 0, v.x * w);
    atomicAdd(ap + 1, v.y * w);
  }
}

// out[i, f] = (relu?) (agg[i, f] + b[f])
template <int OUTD, int RELU>
__global__ __launch_bounds__(256) void k_finalize(const float* __restrict__ agg,
                                                  const float* __restrict__ bias,
                                                  float* __restrict__ out,
                                                  long long total) {
  long long idx = (long long)blockIdx.x * blockDim.x + threadIdx.x;
  if (idx >= total) return;
  float v = agg[idx] + bias[(int)idx & (OUTD - 1)];
  if (RELU) v = fmaxf(v, 0.0f);
  out[idx] = v;
}

// ---------------- host-side layer driver ----------------

template <int OUTD, int RELU>
static void launch_layer(const float* xin, const float* W, const float* b,
                         const int* src, const int* dst,
                         const float* norm, const float* dinv,
                         float* wt, float* h, float* agg, float* out,
                         int E, hipStream_t stream) {
  constexpr int K = 128;
  k_transpose<K, OUTD><<<(K * OUTD + 255) / 256, 256, 0, stream>>>(W, wt);
  k_gemm_wmma<K, OUTD><<<dim3(NN / 16), dim3((OUTD / 16) * 32), 0, stream>>>(xin, wt, h);

  long long tot = (long long)NN * OUTD;
  unsigned nb = (unsigned)((tot + 255) / 256);
  k_self_init<OUTD><<<nb, 256, 0, stream>>>(h, dinv, agg, tot);

  unsigned eb = (unsigned)((E + 7) / 8);  // 8 waves (edges) per 256-thread block
  k_edge_scatter<OUTD><<<eb, 256, 0, stream>>>(src, dst, norm, h, agg, E);

  k_finalize<OUTD, RELU><<<nb, 256, 0, stream>>>(agg, b, out, tot);
}

extern "C" void kernel_launch(void* const* d_in, const int* in_sizes, int n_in,
                              void* d_out, int out_size, void* d_ws, size_t ws_size,
                              hipStream_t stream) {
  const float* x  = (const float*)d_in[0];
  const int*   ei = (const int*)d_in[1];
  const float* W1 = (const float*)d_in[2];
  const float* b1 = (const float*)d_in[3];
  const float* W2 = (const float*)d_in[4];
  const float* b2 = (const float*)d_in[5];
  const float* W3 = (const float*)d_in[6];
  const float* b3 = (const float*)d_in[7];
  const float* W4 = (const float*)d_in[8];
  const float* b4 = (const float*)d_in[9];

  const int E = in_sizes[1] / 2;
  const int* src = ei;
  const int* dst = ei + E;

  // Workspace carve-up (floats); keep 16B alignment for float4 paths.
  float* dinv = (float*)d_ws;                           // NN
  float* norm = dinv + ((NN + 3) & ~3);                 // E
  float* h    = norm + (((size_t)E + 3) & ~(size_t)3);  // NN*128
  float* agg  = h + (size_t)NN * 128;                   // NN*128
  float* xcur = agg + (size_t)NN * 128;                 // NN*128
  float* wt   = xcur + (size_t)NN * 128;                // 128*128

  // GCN normalization (computed once, reused by all 4 layers)
  unsigned nb = (NN + 255) / 256;
  unsigned eb = (unsigned)((E + 255) / 256);
  k_deg_init<<<nb, 256, 0, stream>>>(dinv, NN);
  k_deg_count<<<eb, 256, 0, stream>>>(dst, dinv, E);
  k_dinv<<<nb, 256, 0, stream>>>(dinv, NN);
  k_norm<<<eb, 256, 0, stream>>>(src, dst, dinv, norm, E);

  // Layers 1..3: 128 -> 128 with ReLU; Layer 4: 128 -> 64, no ReLU, into d_out.
  launch_layer<128, 1>(x,    W1, b1, src, dst, norm, dinv, wt, h, agg, xcur, E, stream);
  launch_layer<128, 1>(xcur, W2, b2, src, dst, norm, dinv, wt, h, agg, xcur, E, stream);
  launch_layer<128, 1>(xcur, W3, b3, src, dst, norm, dinv, wt, h, agg, xcur, E, stream);
  launch_layer<64,  0>(xcur, W4, b4, src, dst, norm, dinv, wt, h, agg,
                       (float*)d_out, E, stream);
}